// ElementalLinear_11381663334679
// MI455X (gfx1250) — compile-verified
//
#include <hip/hip_runtime.h>

typedef __attribute__((ext_vector_type(2))) float v2f;
typedef __attribute__((ext_vector_type(4))) float v4f;
typedef __attribute__((ext_vector_type(8))) float v8f;

#define N_PTS 32768
#define E_NUM 4
#define C_DIM 128
#define K_DIM 128
#define PATH_W 0.08838834764831845f /* 1/sqrt(128) */

// LDS B layout: per (step, lane) a contiguous run of 8 v2f fragments (64B),
// padded to 80B (20 floats) row stride => b128-aligned, conflict-free stagger.
#define BROW_F 20                 /* floats per (step,lane) row   */
#define BSTEP_F (32 * BROW_F)     /* floats per step (32 lanes)   */
#define BLDS_BYTES (32 * BSTEP_F * 4) /* 32 steps -> 81920 B      */

// ---------------- bucketize: group point indices by element type ----------------
__global__ void el_init_counts(int* __restrict__ cnt) {
    if (threadIdx.x < E_NUM) cnt[threadIdx.x] = 0;
}

__global__ void el_bucketize(const float* __restrict__ y,
                             int* __restrict__ cnt,
                             int* __restrict__ perm) {
    int n = blockIdx.x * blockDim.x + threadIdx.x;
    if (n >= N_PTS) return;
    const float* yr = y + (size_t)n * E_NUM;
    int e = 0; float best = yr[0];
#pragma unroll
    for (int i = 1; i < E_NUM; ++i) { float v = yr[i]; if (v > best) { best = v; e = i; } }
    int pos = atomicAdd(&cnt[e], 1);
    perm[e * N_PTS + pos] = n;
}

// ---------------- main GEMM kernel ----------------
// Block = 256 threads (8 wave32). blockIdx.y = element bucket e.
// blockIdx.x encodes (l, 128-row group g); worst-case groups per l:
//   l=0:256, l=1:768, l=2:1280, l=3:1792 -> 4096 total.
// Each wave: 16 gathered x-rows times W[e,l] (LDS-staged) = 16x128 output slab,
// 32 K-steps x 8 column tiles of v_wmma_f32_16x16x4_f32, B double-buffered.
extern "C" __global__ __launch_bounds__(256)
void elemental_linear_wmma(const float* __restrict__ x,
                           const float* __restrict__ w,
                           const int* __restrict__ cnt,
                           const int* __restrict__ perm,
                           float* __restrict__ out) {
    extern __shared__ char smem_raw[];
    float* blds = (float*)smem_raw;

    const int e  = blockIdx.y;
    const int bx = blockIdx.x;
    int l, g;
    if      (bx <  256) { l = 0; g = bx; }
    else if (bx < 1024) { l = 1; g = bx - 256; }
    else if (bx < 2304) { l = 2; g = bx - 1024; }
    else                { l = 3; g = bx - 2304; }
    const int m = 2 * l + 1;
    const int s = l * l;

    const int count      = cnt[e];
    const int rows_total = count * m;
    const int row_base   = g * 128;
    if (row_base >= rows_total) return; // uniform per block; before any barrier

    // ---- stage W[e,l] (128x128 f32) into LDS in lane-major WMMA B-fragment order ----
    // v2f at blds[(st*32+lane)*20 + nt*2] = { W[st*4+2*h][nt*16+j], W[st*4+2*h+1][nt*16+j] }
    //   h = lane>>4, j = lane&15 (matches A-side K split: lanes<16 -> K0/K1, lanes>=16 -> K2/K3)
    const float* wbase = w + ((size_t)e * 4 + l) * (C_DIM * K_DIM);
    for (int i = threadIdx.x; i < 32 * 8 * 32; i += 256) {
        int lane_i = i & 31;
        int nt     = (i >> 5) & 7;
        int st     = i >> 8;
        int h = lane_i >> 4, j = lane_i & 15;
        int c = st * 4 + 2 * h;
        int k = nt * 16 + j;
        v2f val;
        val.x = wbase[(size_t)c       * K_DIM + k];
        val.y = wbase[(size_t)(c + 1) * K_DIM + k];
        *(v2f*)&blds[(st * 32 + lane_i) * BROW_F + nt * 2] = val;
    }
    __syncthreads();

    const int lane = threadIdx.x & 31;
    const int wv   = threadIdx.x >> 5;
    const int h    = lane >> 4;
    const int j    = lane & 15;
    const int tile_base = row_base + wv * 16;

    // ---- per-lane A-row pointer (row M = j for this lane's half) ----
    const int* bucket = perm + e * N_PTS;
    const float* arow;
    {
        int R = tile_base + j;
        int p = R / m;
        int mr, n;
        if (p < count) { mr = R - p * m; n = bucket[p]; }
        else           { mr = 0;         n = bucket[0]; } // count>0 guaranteed here
        arow = x + ((size_t)n * 16 + s + mr) * C_DIM + 2 * h;
    }

    v8f acc[8];
    v8f zero = {0.f, 0.f, 0.f, 0.f, 0.f, 0.f, 0.f, 0.f};
#pragma unroll
    for (int nt = 0; nt < 8; ++nt) acc[nt] = zero;

    // ---- K loop: 4x ds_load_b128 per step (all 8 B-frags), double-buffered ----
    const float* bbase = blds + lane * BROW_F;
    v4f c0, c1, c2, c3, p0, p1, p2, p3;
    {
        const v4f* bp = (const v4f*)bbase; // step 0, 80B row => 16B aligned
        c0 = bp[0]; c1 = bp[1]; c2 = bp[2]; c3 = bp[3];
    }
#pragma unroll 4
    for (int st = 0; st < 32; ++st) {
        if (st + 1 < 32) {
            const v4f* bp = (const v4f*)(bbase + (st + 1) * BSTEP_F);
            p0 = bp[0]; p1 = bp[1]; p2 = bp[2]; p3 = bp[3];
        }
        v2f a = *(const v2f*)(arow + st * 4); // 8B-aligned (even element offset)
        v2f b0 = {c0.x, c0.y}, b1 = {c0.z, c0.w};
        v2f b2 = {c1.x, c1.y}, b3 = {c1.z, c1.w};
        v2f b4 = {c2.x, c2.y}, b5 = {c2.z, c2.w};
        v2f b6 = {c3.x, c3.y}, b7 = {c3.z, c3.w};
        acc[0] = __builtin_amdgcn_wmma_f32_16x16x4_f32(false, a, false, b0, (short)0, acc[0], false, false);
        acc[1] = __builtin_amdgcn_wmma_f32_16x16x4_f32(false, a, false, b1, (short)0, acc[1], false, false);
        acc[2] = __builtin_amdgcn_wmma_f32_16x16x4_f32(false, a, false, b2, (short)0, acc[2], false, false);
        acc[3] = __builtin_amdgcn_wmma_f32_16x16x4_f32(false, a, false, b3, (short)0, acc[3], false, false);
        acc[4] = __builtin_amdgcn_wmma_f32_16x16x4_f32(false, a, false, b4, (short)0, acc[4], false, false);
        acc[5] = __builtin_amdgcn_wmma_f32_16x16x4_f32(false, a, false, b5, (short)0, acc[5], false, false);
        acc[6] = __builtin_amdgcn_wmma_f32_16x16x4_f32(false, a, false, b6, (short)0, acc[6], false, false);
        acc[7] = __builtin_amdgcn_wmma_f32_16x16x4_f32(false, a, false, b7, (short)0, acc[7], false, false);
        c0 = p0; c1 = p1; c2 = p2; c3 = p3;
    }

    // ---- store: VGPR r -> (M=r for lanes 0-15, M=r+8 for lanes 16-31), N = nt*16+j ----
#pragma unroll
    for (int r = 0; r < 8; ++r) {
        int R = tile_base + r + 8 * h;
        int p = R / m;
        if (p >= count) continue;
        int mr = R - p * m;
        int n  = bucket[p];
        float* orow = out + ((size_t)n * 16 + s + mr) * K_DIM;
#pragma unroll
        for (int nt = 0; nt < 8; ++nt) {
            orow[nt * 16 + j] = acc[nt][r] * PATH_W;
        }
    }
}

extern "C" void kernel_launch(void* const* d_in, const int* in_sizes, int n_in,
                              void* d_out, int out_size, void* d_ws, size_t ws_size,
                              hipStream_t stream) {
    const float* x = (const float*)d_in[0];
    const float* y = (const float*)d_in[1];
    const float* w = (const float*)d_in[2];
    float* out = (float*)d_out;

    int* iw   = (int*)d_ws;
    int* cnt  = iw;        // 4 counters
    int* perm = iw + 16;   // 4 buckets x N_PTS indices (aligned past counters)

    hipLaunchKernelGGL(el_init_counts, dim3(1), dim3(64), 0, stream, cnt);
    hipLaunchKernelGGL(el_bucketize, dim3(N_PTS / 256), dim3(256), 0, stream, y, cnt, perm);

    dim3 grid(4096, E_NUM);
    hipLaunchKernelGGL(elemental_linear_wmma, grid, dim3(256), BLDS_BYTES, stream,
                       x, w, cnt, perm, out);
}